// PointNetEncoder_22608707846466
// MI455X (gfx1250) — compile-verified
//
#include <hip/hip_runtime.h>
#include <hip/hip_bf16.h>
#include <math.h>

typedef __attribute__((ext_vector_type(2))) float v2f;
typedef __attribute__((ext_vector_type(8))) float v8f;

#define VN_EPS 1e-6f
#define BN_EPS 1e-5f
#define BSZ 16
#define NPT 2048
#define KNN 20
#define COLS (BSZ * 3 * NPT)   // 98304

// ---------------------------------------------------------------------------
// WMMA fp32 GEMM: O[M x cols] = W[M x K] * X[K x cols]
// Block = 512 threads = 16 waves; covers (TWO?32:16) cols x up to 256 rows
// (256-row M-block halves HBM re-reads of X for the big layers).
// W tile staged zero-padded in LDS (KC-chunked, 64 KB); steady-state K loop
// has no guards; <=3-row K tail clamps X row index, zeroed A-frag nullifies.
// ---------------------------------------------------------------------------
template <bool TWO>
__global__ __launch_bounds__(512) void gemm_f32_t(const float* __restrict__ W,
                                                  const float* __restrict__ X,
                                                  float* __restrict__ O,
                                                  int M, int K, int cols) {
  constexpr int KC = 64;
  constexpr int MBLK = 256;
  __shared__ float Wl[MBLK * KC];           // 64 KB
  const int wave = threadIdx.x >> 5;        // 0..15
  const int lane = threadIdx.x & 31;
  const int colbase = blockIdx.x * (TWO ? 32 : 16);
  const int mblock = blockIdx.y * MBLK;
  const int l16 = lane & 15;
  const int khalf = (lane >> 4) << 1;       // 0 or 2
  const int col = colbase + l16;
  const int rowbase = mblock + wave * 16;
  const bool active = (rowbase < M);        // wave-uniform
  const float* wl0 = &Wl[(wave * 16 + l16) * KC + khalf];
  v8f acc0 = {0.f, 0.f, 0.f, 0.f, 0.f, 0.f, 0.f, 0.f};
  v8f acc1 = {0.f, 0.f, 0.f, 0.f, 0.f, 0.f, 0.f, 0.f};

  for (int kc = 0; kc < K; kc += KC) {
    const int klen = (K - kc < KC) ? (K - kc) : KC;
    __syncthreads();
    // cooperative, coalesced, zero-padded W stage
    for (int t = threadIdx.x; t < MBLK * KC; t += 512) {
      const int r = t >> 6;                 // t / KC
      const int c = t & (KC - 1);           // t % KC
      const int gr = mblock + r;
      Wl[t] = (gr < M && c < klen) ? W[(size_t)gr * K + (kc + c)] : 0.f;
    }
    __syncthreads();
    if (active) {
      const float* wl = wl0;
      const float* xp = X + (size_t)(kc + khalf) * cols + col;
      int k0 = 0;
#pragma unroll 4
      for (; k0 + 4 <= klen; k0 += 4) {
        v2f a, b0;
        a.x = wl[0]; a.y = wl[1];           // ds_load_2addr
        b0.x = xp[0]; b0.y = xp[cols];
        if (TWO) {
          v2f b1;
          b1.x = xp[16]; b1.y = xp[cols + 16];
          acc1 = __builtin_amdgcn_wmma_f32_16x16x4_f32(false, a, false, b1,
                                                       (short)0, acc1, false, false);
        }
        acc0 = __builtin_amdgcn_wmma_f32_16x16x4_f32(false, a, false, b0,
                                                     (short)0, acc0, false, false);
        wl += 4;
        xp += (size_t)4 * cols;
      }
      if (k0 < klen) {                      // K tail (<=3 rows), once per GEMM
        const int ka = k0 + khalf;
        const int kb = ka + 1;
        const int kca = (ka < klen ? ka : klen - 1);   // clamp: valid row,
        const int kcb = (kb < klen ? kb : klen - 1);   // A-frag is 0 there
        v2f a, b0;
        a.x = wl[0]; a.y = wl[1];           // zero-padded in LDS
        const float* xb = X + (size_t)kc * cols + col;
        b0.x = xb[(size_t)kca * cols]; b0.y = xb[(size_t)kcb * cols];
        if (TWO) {
          v2f b1;
          b1.x = xb[(size_t)kca * cols + 16]; b1.y = xb[(size_t)kcb * cols + 16];
          acc1 = __builtin_amdgcn_wmma_f32_16x16x4_f32(false, a, false, b1,
                                                       (short)0, acc1, false, false);
        }
        acc0 = __builtin_amdgcn_wmma_f32_16x16x4_f32(false, a, false, b0,
                                                     (short)0, acc0, false, false);
      }
    }
  }
  if (active) {
    const int rhalf = (lane >> 4) * 8;
    for (int v = 0; v < 8; ++v) {
      const int row = rowbase + v + rhalf;
      if (row < M) {
        O[(size_t)row * cols + col] = acc0[v];
        if (TWO) O[(size_t)row * cols + col + 16] = acc1[v];
      }
    }
  }
}

// ---------------------------------------------------------------------------
// kNN: one block per point; pairwise "pd" in LDS, 20 argmax passes (ties ->
// lowest index, matching jax.lax.top_k).
// ---------------------------------------------------------------------------
__global__ __launch_bounds__(256) void knn_kernel(const float* __restrict__ x,
                                                  int* __restrict__ idx) {
  __shared__ float pd[NPT];
  __shared__ float rmax[256];
  __shared__ int rarg[256];
  const int b = blockIdx.x / NPT, n0 = blockIdx.x % NPT;
  const float* xb = x + (size_t)b * 3 * NPT;
  const float cx = xb[n0], cy = xb[NPT + n0], cz = xb[2 * NPT + n0];
  const float c2 = cx * cx + cy * cy + cz * cz;
  for (int n = threadIdx.x; n < NPT; n += blockDim.x) {
    const float px = xb[n], py = xb[NPT + n], pz = xb[2 * NPT + n];
    pd[n] = 2.f * (cx * px + cy * py + cz * pz) - c2 - (px * px + py * py + pz * pz);
  }
  __syncthreads();
  for (int j = 0; j < KNN; ++j) {
    float best = -3.0e38f; int barg = 0x7fffffff;
    for (int n = threadIdx.x; n < NPT; n += blockDim.x) {
      const float d = pd[n];
      if (d > best) { best = d; barg = n; }
    }
    rmax[threadIdx.x] = best; rarg[threadIdx.x] = barg;
    __syncthreads();
    for (int s = 128; s > 0; s >>= 1) {
      if (threadIdx.x < s) {
        const float o = rmax[threadIdx.x + s]; const int oa = rarg[threadIdx.x + s];
        if (o > rmax[threadIdx.x] ||
            (o == rmax[threadIdx.x] && oa < rarg[threadIdx.x])) {
          rmax[threadIdx.x] = o; rarg[threadIdx.x] = oa;
        }
      }
      __syncthreads();
    }
    if (threadIdx.x == 0) {
      idx[(size_t)blockIdx.x * KNN + j] = rarg[0];
      pd[rarg[0]] = -3.3e38f;
    }
    __syncthreads();
  }
}

// ---------------------------------------------------------------------------
// conv_pos (fused; avoids materializing P/D over N*k):
// stats pass recomputes P=Wf@feat and accumulates sum/sumsq of ||P|| per chan.
// ---------------------------------------------------------------------------
__global__ __launch_bounds__(256) void convpos_stats(const float* __restrict__ x,
                                                     const int* __restrict__ idx,
                                                     const float* __restrict__ Wf,
                                                     float* __restrict__ stats) {
  const int item = blockIdx.x * 256 + threadIdx.x;   // (b*N+n)*k + kk  (exact)
  const int kk = item % KNN;
  const int bn = item / KNN;
  const int n = bn % NPT, b = bn / NPT;
  const float* xb = x + (size_t)b * 3 * NPT;
  const int nb = idx[(size_t)bn * KNN + kk];
  const float cx = xb[n], cy = xb[NPT + n], cz = xb[2 * NPT + n];
  const float px = xb[nb], py = xb[NPT + nb], pz = xb[2 * NPT + nb];
  float f[3][3];
  f[0][0] = px - cx; f[0][1] = py - cy; f[0][2] = pz - cz;
  f[1][0] = cx;      f[1][1] = cy;      f[1][2] = cz;
  f[2][0] = py * cz - pz * cy; f[2][1] = pz * cx - px * cz; f[2][2] = px * cy - py * cx;
  float nrm[21];
  for (int o = 0; o < 21; ++o) {
    float s = 0.f;
    const float w0 = Wf[o * 3 + 0], w1 = Wf[o * 3 + 1], w2 = Wf[o * 3 + 2];
    for (int d = 0; d < 3; ++d) {
      const float p = w0 * f[0][d] + w1 * f[1][d] + w2 * f[2][d];
      s += p * p;
    }
    nrm[o] = sqrtf(s) + VN_EPS;
  }
  __shared__ float rs[256], rq[256];
  for (int o = 0; o < 21; ++o) {
    rs[threadIdx.x] = nrm[o]; rq[threadIdx.x] = nrm[o] * nrm[o];
    __syncthreads();
    for (int s = 128; s > 0; s >>= 1) {
      if (threadIdx.x < s) { rs[threadIdx.x] += rs[threadIdx.x + s];
                             rq[threadIdx.x] += rq[threadIdx.x + s]; }
      __syncthreads();
    }
    if (threadIdx.x == 0) { atomicAdd(&stats[o], rs[0]); atomicAdd(&stats[21 + o], rq[0]); }
    __syncthreads();
  }
}

// apply pass: one thread per (b,n,o); recompute feat/P/D per k, BN + VN-ReLU,
// mean over k -> H1[o][(b*3+d)*N+n]
__global__ __launch_bounds__(256) void convpos_apply(const float* __restrict__ x,
                                                     const int* __restrict__ idx,
                                                     const float* __restrict__ Wf,
                                                     const float* __restrict__ Wd,
                                                     const float* __restrict__ ss,
                                                     float* __restrict__ H1) {
  const int item = blockIdx.x * 256 + threadIdx.x;
  if (item >= BSZ * NPT * 21) return;
  const int o = item % 21;
  const int bn = item / 21;
  const int n = bn % NPT, b = bn / NPT;
  const float* xb = x + (size_t)b * 3 * NPT;
  const float cx = xb[n], cy = xb[NPT + n], cz = xb[2 * NPT + n];
  const float scale = ss[o], shift = ss[21 + o];
  const float wf0 = Wf[o * 3], wf1 = Wf[o * 3 + 1], wf2 = Wf[o * 3 + 2];
  const float wd0 = Wd[o * 3], wd1 = Wd[o * 3 + 1], wd2 = Wd[o * 3 + 2];
  float a0 = 0.f, a1 = 0.f, a2 = 0.f;
  for (int kk = 0; kk < KNN; ++kk) {
    const int nb = idx[(size_t)bn * KNN + kk];
    const float px = xb[nb], py = xb[NPT + nb], pz = xb[2 * NPT + nb];
    const float e0 = px - cx, e1 = py - cy, e2 = pz - cz;
    const float r0 = py * cz - pz * cy, r1 = pz * cx - px * cz, r2 = px * cy - py * cx;
    float P0 = wf0 * e0 + wf1 * cx + wf2 * r0;
    float P1 = wf0 * e1 + wf1 * cy + wf2 * r1;
    float P2 = wf0 * e2 + wf1 * cz + wf2 * r2;
    const float D0 = wd0 * e0 + wd1 * cx + wd2 * r0;
    const float D1 = wd0 * e1 + wd1 * cy + wd2 * r1;
    const float D2 = wd0 * e2 + wd1 * cz + wd2 * r2;
    const float nrm = sqrtf(P0 * P0 + P1 * P1 + P2 * P2) + VN_EPS;
    const float t = (nrm * scale + shift) / nrm;
    P0 *= t; P1 *= t; P2 *= t;
    const float dot = P0 * D0 + P1 * D1 + P2 * D2;
    if (dot < 0.f) {
      const float fct = dot / (D0 * D0 + D1 * D1 + D2 * D2 + VN_EPS);
      P0 -= fct * D0; P1 -= fct * D1; P2 -= fct * D2;
    }
    a0 += P0; a1 += P1; a2 += P2;
  }
  const float inv = 1.f / KNN;
  const size_t cb = ((size_t)b * 3) * NPT + n;
  H1[(size_t)o * COLS + cb]            = a0 * inv;
  H1[(size_t)o * COLS + cb + NPT]      = a1 * inv;
  H1[(size_t)o * COLS + cb + 2 * NPT]  = a2 * inv;
}

// ---------------------------------------------------------------------------
// Generic VN-BN stats over P: per channel c, sum & sumsq of ||P(b,c,:,n)||.
// ---------------------------------------------------------------------------
__global__ __launch_bounds__(256) void vn_stats(const float* __restrict__ P,
                                                float* __restrict__ stats,
                                                int C, int Ns, int B) {
  const int c = blockIdx.y;
  const int i = blockIdx.x * 256 + threadIdx.x;
  const int cols = B * 3 * Ns;
  float v = 0.f, v2 = 0.f;
  if (i < B * Ns) {
    const int b = i / Ns, n = i % Ns;
    const float* row = P + (size_t)c * cols;
    const size_t col = ((size_t)b * 3) * Ns + n;
    const float p0 = row[col], p1 = row[col + Ns], p2 = row[col + 2 * Ns];
    const float nrm = sqrtf(p0 * p0 + p1 * p1 + p2 * p2) + VN_EPS;
    v = nrm; v2 = nrm * nrm;
  }
  __shared__ float rs[256], rq[256];
  rs[threadIdx.x] = v; rq[threadIdx.x] = v2;
  __syncthreads();
  for (int s = 128; s > 0; s >>= 1) {
    if (threadIdx.x < s) { rs[threadIdx.x] += rs[threadIdx.x + s];
                           rq[threadIdx.x] += rq[threadIdx.x + s]; }
    __syncthreads();
  }
  if (threadIdx.x == 0) { atomicAdd(&stats[c], rs[0]); atomicAdd(&stats[C + c], rq[0]); }
}

__global__ void vn_finalize(const float* __restrict__ stats,
                            const float* __restrict__ g,
                            const float* __restrict__ bb,
                            float* __restrict__ ss, int C, float count) {
  const int c = blockIdx.x * blockDim.x + threadIdx.x;
  if (c >= C) return;
  const float mu = stats[c] / count;
  const float var = stats[C + c] / count - mu * mu;
  const float sc = g[c] * rsqrtf(var + BN_EPS);
  ss[c] = sc; ss[C + c] = bb[c] - mu * sc;
}

// BN + (optional) VN-LeakyReLU(0); in-place over P.
__global__ __launch_bounds__(256) void vn_combine(float* __restrict__ P,
                                                  const float* __restrict__ D,
                                                  const float* __restrict__ ss,
                                                  int C, int Ns, int B, int relu) {
  const int c = blockIdx.y;
  const int i = blockIdx.x * 256 + threadIdx.x;
  if (i >= B * Ns) return;
  const int b = i / Ns, n = i % Ns;
  const size_t cols = (size_t)B * 3 * Ns;
  const size_t col = ((size_t)b * 3) * Ns + n;
  float* pr = P + (size_t)c * cols + col;
  float p0 = pr[0], p1 = pr[Ns], p2 = pr[2 * Ns];
  const float nrm = sqrtf(p0 * p0 + p1 * p1 + p2 * p2) + VN_EPS;
  const float t = (nrm * ss[c] + ss[C + c]) / nrm;
  p0 *= t; p1 *= t; p2 *= t;
  if (relu) {
    const float* dr = D + (size_t)c * cols + col;
    const float d0 = dr[0], d1 = dr[Ns], d2 = dr[2 * Ns];
    const float dot = p0 * d0 + p1 * d1 + p2 * d2;
    if (dot < 0.f) {
      const float f = dot / (d0 * d0 + d1 * d1 + d2 * d2 + VN_EPS);
      p0 -= f * d0; p1 -= f * d1; p2 -= f * d2;
    }
  }
  pr[0] = p0; pr[Ns] = p1; pr[2 * Ns] = p2;
}

// mean over n -> small tensor OutSmall[c][(b*3+d)]
__global__ __launch_bounds__(256) void row_mean(const float* __restrict__ X,
                                                float* __restrict__ OutSmall,
                                                int Ns, int B) {
  const int bd = blockIdx.x % (B * 3);
  const int c = blockIdx.x / (B * 3);
  const float* row = X + ((size_t)c * (B * 3) + bd) * Ns;
  float s = 0.f;
  for (int n = threadIdx.x; n < Ns; n += blockDim.x) s += row[n];
  __shared__ float r[256];
  r[threadIdx.x] = s; __syncthreads();
  for (int st = 128; st > 0; st >>= 1) {
    if (threadIdx.x < st) r[threadIdx.x] += r[threadIdx.x + st];
    __syncthreads();
  }
  if (threadIdx.x == 0) OutSmall[(size_t)c * (B * 3) + bd] = r[0] / (float)Ns;
}

// rows [C..2C): broadcast of per-(c,b,d) mean over n (the h-concat).
__global__ __launch_bounds__(256) void row_mean_bcast(float* __restrict__ X,
                                                      int C, int Ns, int B) {
  const int bd = blockIdx.x % (B * 3);
  const int c = blockIdx.x / (B * 3);
  const size_t stride = (size_t)B * 3 * Ns;
  const float* row = X + (size_t)c * stride + (size_t)bd * Ns;
  float s = 0.f;
  for (int n = threadIdx.x; n < Ns; n += blockDim.x) s += row[n];
  __shared__ float r[256];
  r[threadIdx.x] = s; __syncthreads();
  for (int st = 128; st > 0; st >>= 1) {
    if (threadIdx.x < st) r[threadIdx.x] += r[threadIdx.x + st];
    __syncthreads();
  }
  const float m = r[0] / (float)Ns;
  __syncthreads();
  float* orow = X + (size_t)(c + C) * stride + (size_t)bd * Ns;
  for (int n = threadIdx.x; n < Ns; n += blockDim.x) orow[n] = m;
}

// broadcast STN output s (C x (B*3)) over n into HC rows [C..2C)
__global__ __launch_bounds__(256) void bcast_s(const float* __restrict__ S,
                                               float* __restrict__ HC,
                                               int C, int Ns, int B) {
  const int i = blockIdx.x * 256 + threadIdx.x;
  const int total = C * B * 3 * Ns;
  if (i >= total) return;
  const int n = i % Ns;
  const int rest = i / Ns;
  const int bd = rest % (B * 3);
  const int c = rest / (B * 3);
  HC[((size_t)(C + c) * (B * 3) + bd) * Ns + n] = S[(size_t)c * (B * 3) + bd];
}

// z0 transpose: Z[o][(b*3+d)*Ns+n] -> out[((b*3+d)*3+o)*Ns+n]
__global__ __launch_bounds__(256) void z0_transpose(const float* __restrict__ Z,
                                                    float* __restrict__ out,
                                                    int B, int Ns) {
  const int i = blockIdx.x * 256 + threadIdx.x;
  const int total = 9 * B * Ns;
  if (i >= total) return;
  const int n = i % Ns;
  const int r = i / Ns;
  const int o = r % 3;
  const int bd = r / 3;
  out[i] = Z[((size_t)o * (B * 3) + bd) * Ns + n];
}

// final: xs[b,i,kk,n] = sum_j h[i][(b*3+j)*N+n] * Z[kk][(b*3+j)*N+n]; max over n
__global__ __launch_bounds__(256) void final_max(const float* __restrict__ H,
                                                 const float* __restrict__ Z,
                                                 float* __restrict__ out,
                                                 int B, int Ns) {
  const int i = blockIdx.x % 682;
  const int b = blockIdx.x / 682;
  const size_t cols = (size_t)B * 3 * Ns;
  const float* h0 = H + (size_t)i * cols + (size_t)(b * 3) * Ns;
  const size_t cb0 = (size_t)(b * 3) * Ns;
  float m0 = -3.0e38f, m1 = -3.0e38f, m2 = -3.0e38f;
  for (int n = threadIdx.x; n < Ns; n += blockDim.x) {
    const float h_0 = h0[n], h_1 = h0[Ns + n], h_2 = h0[2 * Ns + n];
    const size_t cb = cb0 + n;
    const float x0 = h_0 * Z[cb] + h_1 * Z[cb + Ns] + h_2 * Z[cb + 2 * Ns];
    const float x1 = h_0 * Z[cols + cb] + h_1 * Z[cols + cb + Ns] + h_2 * Z[cols + cb + 2 * Ns];
    const float x2 = h_0 * Z[2 * cols + cb] + h_1 * Z[2 * cols + cb + Ns] + h_2 * Z[2 * cols + cb + 2 * Ns];
    m0 = fmaxf(m0, x0); m1 = fmaxf(m1, x1); m2 = fmaxf(m2, x2);
  }
  __shared__ float r0[256], r1[256], r2[256];
  r0[threadIdx.x] = m0; r1[threadIdx.x] = m1; r2[threadIdx.x] = m2;
  __syncthreads();
  for (int st = 128; st > 0; st >>= 1) {
    if (threadIdx.x < st) {
      r0[threadIdx.x] = fmaxf(r0[threadIdx.x], r0[threadIdx.x + st]);
      r1[threadIdx.x] = fmaxf(r1[threadIdx.x], r1[threadIdx.x + st]);
      r2[threadIdx.x] = fmaxf(r2[threadIdx.x], r2[threadIdx.x + st]);
    }
    __syncthreads();
  }
  if (threadIdx.x == 0) {
    out[(size_t)b * 2046 + i * 3 + 0] = r0[0];
    out[(size_t)b * 2046 + i * 3 + 1] = r1[0];
    out[(size_t)b * 2046 + i * 3 + 2] = r2[0];
  }
}

__global__ void zero_kernel(float* p, int n) {
  for (int i = threadIdx.x; i < n; i += blockDim.x) p[i] = 0.f;
}

// ---------------------------------------------------------------------------
extern "C" void kernel_launch(void* const* d_in, const int* in_sizes, int n_in,
                              void* d_out, int out_size, void* d_ws, size_t ws_size,
                              hipStream_t stream) {
  (void)in_sizes; (void)n_in; (void)out_size; (void)ws_size;
  const int B = BSZ, N = NPT;
  const float* x      = (const float*)d_in[0];
  const float* cpWf   = (const float*)d_in[1];
  const float* cpWd   = (const float*)d_in[2];
  const float* cpg    = (const float*)d_in[3];
  const float* cpb    = (const float*)d_in[4];
  const float* c1Wf   = (const float*)d_in[5];
  const float* c1Wd   = (const float*)d_in[6];
  const float* c1g    = (const float*)d_in[7];
  const float* c1b    = (const float*)d_in[8];
  const float* s1Wf   = (const float*)d_in[9];
  const float* s1Wd   = (const float*)d_in[10];
  const float* s1g    = (const float*)d_in[11];
  const float* s1b    = (const float*)d_in[12];
  const float* s2Wf   = (const float*)d_in[13];
  const float* s2Wd   = (const float*)d_in[14];
  const float* s2g    = (const float*)d_in[15];
  const float* s2b    = (const float*)d_in[16];
  const float* s3Wf   = (const float*)d_in[17];
  const float* s3Wd   = (const float*)d_in[18];
  const float* s3g    = (const float*)d_in[19];
  const float* s3b    = (const float*)d_in[20];
  const float* f1Wf   = (const float*)d_in[21];
  const float* f1Wd   = (const float*)d_in[22];
  const float* f1g    = (const float*)d_in[23];
  const float* f1b    = (const float*)d_in[24];
  const float* f2Wf   = (const float*)d_in[25];
  const float* f2Wd   = (const float*)d_in[26];
  const float* f2g    = (const float*)d_in[27];
  const float* f2b    = (const float*)d_in[28];
  const float* fc3W   = (const float*)d_in[29];
  const float* c2Wf   = (const float*)d_in[30];
  const float* c2Wd   = (const float*)d_in[31];
  const float* c2g    = (const float*)d_in[32];
  const float* c2b    = (const float*)d_in[33];
  const float* c3W    = (const float*)d_in[34];
  const float* bn3g   = (const float*)d_in[35];
  const float* bn3b   = (const float*)d_in[36];
  const float* v1Wf   = (const float*)d_in[37];
  const float* v1Wd   = (const float*)d_in[38];
  const float* v1g    = (const float*)d_in[39];
  const float* v1b    = (const float*)d_in[40];
  const float* v2Wf   = (const float*)d_in[41];
  const float* v2Wd   = (const float*)d_in[42];
  const float* v2g    = (const float*)d_in[43];
  const float* v2b    = (const float*)d_in[44];
  const float* slW    = (const float*)d_in[45];
  float* out = (float*)d_out;

  // workspace layout (floats)
  float* ws = (float*)d_ws;
  size_t off = 0;
  int* idxb = (int*)ws;            off += (size_t)B * N * KNN;       // kNN indices
  float* stats = ws + off;         off += 682;
  float* ss    = ws + off;         off += 682;
  float* SM    = ws + off;         off += 341 * 48;
  float* F1    = ws + off;         off += 170 * 48;
  float* F1D   = ws + off;         off += 170 * 48;
  float* F2    = ws + off;         off += 85 * 48;
  float* F2D   = ws + off;         off += 85 * 48;
  float* SOUT  = ws + off;         off += 21 * 48;
  float* R1    = ws + off;         off += (size_t)21  * COLS;        // H1 / S1
  float* R2    = ws + off;         off += (size_t)21  * COLS;        // h (conv1 out)
  float* R3    = ws + off;         off += (size_t)42  * COLS;        // S2 / h-concat
  float* R4    = ws + off;         off += (size_t)341 * COLS;        // S3 / conv2out / Z1
  float* R5    = ws + off;         off += (size_t)682 * COLS;        // h3 + mean concat
  float* R6    = ws + off;         off += (size_t)341 * COLS;        // D scratch
  float* R7    = ws + off;         off += (size_t)170 * COLS;        // Z2
  float* R8    = ws + off;         off += (size_t)3   * COLS;        // z0 (pre-transpose)

  auto launch_gemm = [&](const float* Wm, const float* Xm, float* Om,
                         int M, int K, int cols) {
    if (cols % 32 == 0) {
      gemm_f32_t<true><<<dim3(cols / 32, (M + 255) / 256), 512, 0, stream>>>(
          Wm, Xm, Om, M, K, cols);
    } else {
      gemm_f32_t<false><<<dim3(cols / 16, (M + 255) / 256), 512, 0, stream>>>(
          Wm, Xm, Om, M, K, cols);
    }
  };

  auto vn_layer = [&](const float* Wf, const float* Wd, const float* g,
                      const float* bb, const float* X, float* P, float* D,
                      int Cin, int Cout, int Ns) {
    const int cols = B * 3 * Ns;
    launch_gemm(Wf, X, P, Cout, Cin, cols);
    launch_gemm(Wd, X, D, Cout, Cin, cols);
    zero_kernel<<<1, 256, 0, stream>>>(stats, 2 * Cout);
    dim3 sg((B * Ns + 255) / 256, Cout);
    vn_stats<<<sg, 256, 0, stream>>>(P, stats, Cout, Ns, B);
    vn_finalize<<<(Cout + 255) / 256, 256, 0, stream>>>(stats, g, bb, ss, Cout,
                                                        (float)(B * Ns));
    vn_combine<<<sg, 256, 0, stream>>>(P, D, ss, Cout, Ns, B, 1);
  };

  // 1) kNN
  knn_kernel<<<B * N, 256, 0, stream>>>(x, idxb);

  // 2) conv_pos (fused over N*k) -> R1 (21)
  zero_kernel<<<1, 256, 0, stream>>>(stats, 42);
  convpos_stats<<<(B * N * KNN) / 256, 256, 0, stream>>>(x, idxb, cpWf, stats);
  vn_finalize<<<1, 256, 0, stream>>>(stats, cpg, cpb, ss, 21, (float)(B * N * KNN));
  convpos_apply<<<(B * N * 21 + 255) / 256, 256, 0, stream>>>(x, idxb, cpWf, cpWd,
                                                              ss, R1);

  // 3) conv1: R1 -> R2 (h, 21)
  vn_layer(c1Wf, c1Wd, c1g, c1b, R1, R2, R6, 21, 21, N);

  // 4) STN convs: R2 -> R1 -> R3 -> R4
  vn_layer(s1Wf, s1Wd, s1g, s1b, R2, R1, R6, 21, 21, N);
  vn_layer(s2Wf, s2Wd, s2g, s2b, R1, R3, R6, 21, 42, N);
  vn_layer(s3Wf, s3Wd, s3g, s3b, R3, R4, R6, 42, 341, N);
  row_mean<<<341 * (B * 3), 256, 0, stream>>>(R4, SM, N, B);

  // 5) STN FCs (Ns = 1, cols = 48)
  vn_layer(f1Wf, f1Wd, f1g, f1b, SM, F1, F1D, 341, 170, 1);
  vn_layer(f2Wf, f2Wd, f2g, f2b, F1, F2, F2D, 170, 85, 1);
  launch_gemm(fc3W, F2, SOUT, 21, 85, 48);

  // 6) concat [h, broadcast(s)] -> R3 (42)
  hipMemcpyAsync(R3, R2, (size_t)21 * COLS * sizeof(float),
                 hipMemcpyDeviceToDevice, stream);
  bcast_s<<<(21 * B * 3 * N + 255) / 256, 256, 0, stream>>>(SOUT, R3, 21, N, B);

  // 7) conv2: R3 -> R4 (42)
  vn_layer(c2Wf, c2Wd, c2g, c2b, R3, R4, R6, 42, 42, N);

  // 8) conv3 + bn3 -> R5 rows [0,341)
  launch_gemm(c3W, R4, R5, 341, 42, COLS);
  zero_kernel<<<1, 256, 0, stream>>>(stats, 2 * 341);
  {
    dim3 sg((B * N + 255) / 256, 341);
    vn_stats<<<sg, 256, 0, stream>>>(R5, stats, 341, N, B);
    vn_finalize<<<2, 256, 0, stream>>>(stats, bn3g, bn3b, ss, 341, (float)(B * N));
    vn_combine<<<sg, 256, 0, stream>>>(R5, R5, ss, 341, N, B, 0);
  }
  // 9) concat with per-point mean -> R5 rows [341,682)
  row_mean_bcast<<<341 * (B * 3), 256, 0, stream>>>(R5, 341, N, B);

  // 10) std head
  vn_layer(v1Wf, v1Wd, v1g, v1b, R5, R4, R6, 682, 341, N);
  vn_layer(v2Wf, v2Wd, v2g, v2b, R4, R7, R6, 341, 170, N);
  launch_gemm(slW, R7, R8, 3, 170, COLS);

  // 11) outputs: z0 (second output) + max-pooled features (first output)
  z0_transpose<<<(9 * B * N + 255) / 256, 256, 0, stream>>>(R8, out + (size_t)B * 2046,
                                                            B, N);
  final_max<<<B * 682, 256, 0, stream>>>(R5, R8, out, B, N);
}